// MeshTransformer_70196945485984
// MI455X (gfx1250) — compile-verified
//
#include <hip/hip_runtime.h>
#include <hip/hip_bf16.h>
#include <math.h>
#include <stdint.h>

typedef float v2f __attribute__((ext_vector_type(2)));
typedef float v8f __attribute__((ext_vector_type(8)));
// pointee type for the async-copy builtin, exactly as clang's diagnostic prints it
typedef int v4i_ __attribute__((__vector_size__(4 * sizeof(int))));

#define S_     1024
#define FEAT_  64
#define D_     512
#define H_     8
#define HD_    64
#define L_     2
#define DFF_   2048
#define NC_    10
#define DB_    128
#define NFREQ_ 85
#define EPS_   1e-5f
// Tile row pitch (floats) for [row][k] LDS tiles. 20*lm mod 64 gives 16
// distinct banks, all == 0 (mod 4); the upper half-wave reads kr+2 ->
// banks == 2 (mod 4) -> the wave's 64 dwords of a b64 fragment load are
// fully conflict-free. 20*4 = 80 bytes keeps 16B alignment for b128 copies.
#define AP     20
#define TILE_F (64 * AP)            // floats per tile buffer

// ---------------- CDNA5 async global->LDS copy (guarded) -----------------
#if defined(__has_builtin)
# if __has_builtin(__builtin_amdgcn_global_load_async_to_lds_b128)
#  define HAVE_ASYNC 1
# endif
#endif
#ifndef HAVE_ASYNC
# define HAVE_ASYNC 0
#endif

// copy 16 bytes global -> LDS (async when available, else via VGPRs)
__device__ __forceinline__ void cp16(const float* g, float* l) {
#if HAVE_ASYNC
  __builtin_amdgcn_global_load_async_to_lds_b128(
      (__attribute__((address_space(1))) v4i_*)(uintptr_t)g,
      (__attribute__((address_space(3))) v4i_*)(uintptr_t)l,
      0, 0);
#else
  *(float4*)l = *(const float4*)g;
#endif
}

__device__ __forceinline__ void async_drain() {
#if HAVE_ASYNC
# if defined(__has_builtin) && __has_builtin(__builtin_amdgcn_s_wait_asynccnt)
  __builtin_amdgcn_s_wait_asynccnt(0);
# else
  asm volatile("s_wait_asynccnt 0" ::: "memory");
# endif
#endif
}

// ---------------- 16 K-steps of WMMA on one [row][k] tile pair -----------
__device__ __forceinline__ void wmma_tile16(const float* AsB, const float* BsB,
                                            int wm, int wn, int lh, int lm,
                                            v8f acc[2][2]) {
  #pragma unroll
  for (int kk = 0; kk < 16; kk += 4) {
    const int kr = kk + 2 * lh;   // lanes 0-15: K=kk,kk+1; lanes 16-31: kk+2,kk+3
    v2f af0 = *(const v2f*)(AsB + (wm + lm) * AP + kr);
    v2f af1 = *(const v2f*)(AsB + (wm + 16 + lm) * AP + kr);
    v2f bf0 = *(const v2f*)(BsB + (wn + lm) * AP + kr);
    v2f bf1 = *(const v2f*)(BsB + (wn + 16 + lm) * AP + kr);
    acc[0][0] = __builtin_amdgcn_wmma_f32_16x16x4_f32(false, af0, false, bf0,
                                                      (short)0, acc[0][0], false, false);
    acc[0][1] = __builtin_amdgcn_wmma_f32_16x16x4_f32(false, af0, false, bf1,
                                                      (short)0, acc[0][1], false, false);
    acc[1][0] = __builtin_amdgcn_wmma_f32_16x16x4_f32(false, af1, false, bf0,
                                                      (short)0, acc[1][0], false, false);
    acc[1][1] = __builtin_amdgcn_wmma_f32_16x16x4_f32(false, af1, false, bf1,
                                                      (short)0, acc[1][1], false, false);
  }
}

// ------------------------------------------------------------------
// x = features @ in_w + in_b + positional_encoding
// ------------------------------------------------------------------
__global__ void __launch_bounds__(256)
k_input_pe(const float* __restrict__ feats, const float* __restrict__ pos,
           const float* __restrict__ fb, const float* __restrict__ in_w,
           const float* __restrict__ in_b, float* __restrict__ x) {
  int idx = blockIdx.x * 256 + threadIdx.x;       // S*D threads
  int s = idx >> 9, d = idx & (D_ - 1);
  float acc = in_b[d];
  const float* fr = feats + s * FEAT_;
  #pragma unroll 4
  for (int f = 0; f < FEAT_; ++f) acc = fmaf(fr[f], in_w[f * D_ + d], acc);
  float pe = 0.f;
  if (d < 6 * NFREQ_) {
    int axis = d / (2 * NFREQ_);
    int rem = d - axis * 2 * NFREQ_;
    float p = pos[s * 3 + axis];
    pe = (rem < NFREQ_) ? sinf(p * fb[rem]) : cosf(p * fb[rem - NFREQ_]);
  }
  x[idx] = acc + pe;
}

// ------------------------------------------------------------------
// pairwise distances
// ------------------------------------------------------------------
__global__ void __launch_bounds__(256)
k_dist(const float* __restrict__ pos, float* __restrict__ dist) {
  int idx = blockIdx.x * 256 + threadIdx.x;       // S*S threads
  int i = idx >> 10, j = idx & (S_ - 1);
  float dx = pos[i * 3 + 0] - pos[j * 3 + 0];
  float dy = pos[i * 3 + 1] - pos[j * 3 + 1];
  float dz = pos[i * 3 + 2] - pos[j * 3 + 2];
  float sq = dx * dx + dy * dy + dz * dz;
  dist[idx] = sq > 0.f ? sqrtf(sq) : 0.f;
}

// ------------------------------------------------------------------
// distance-bias MLP: att[h,i,j] = sum_c relu(dist*w1[c]+b1[c])*w2[c,h] + b2[h]
// (pre-fills the score buffer; the QK^T GEMM accumulates onto it)
// ------------------------------------------------------------------
__global__ void __launch_bounds__(256)
k_bias(const float* __restrict__ dist, const float* __restrict__ w1,
       const float* __restrict__ b1, const float* __restrict__ w2,
       const float* __restrict__ b2, float* __restrict__ att) {
  __shared__ float w1s[DB_], b1s[DB_], w2s[DB_ * H_], b2s[H_];
  int t = threadIdx.x;
  if (t < DB_) { w1s[t] = w1[t]; b1s[t] = b1[t]; }
  if (t < H_) b2s[t] = b2[t];
  for (int c = t; c < DB_ * H_; c += 256) w2s[c] = w2[c];
  __syncthreads();
  long long idx = (long long)blockIdx.x * 256 + t;
  float dv = dist[idx];
  float acc[H_];
  #pragma unroll
  for (int h = 0; h < H_; ++h) acc[h] = b2s[h];
  for (int c = 0; c < DB_; ++c) {
    float hb = fmaxf(fmaf(dv, w1s[c], b1s[c]), 0.f);
    #pragma unroll
    for (int h = 0; h < H_; ++h) acc[h] = fmaf(hb, w2s[c * H_ + h], acc[h]);
  }
  #pragma unroll
  for (int h = 0; h < H_; ++h) att[(long long)h * S_ * S_ + idx] = acc[h];
}

// ------------------------------------------------------------------
// C[m,n] = act(A @ B + bias) via V_WMMA_F32_16X16X4_F32.
// 128 threads (4 waves), 64x64 tile, BK=16, double-buffered LDS.
// A tile rows copied async (direct [m][k]); B tile transposed via VGPRs.
// ------------------------------------------------------------------
template <bool RELU, bool HAS_BIAS>
__global__ void __launch_bounds__(128)
k_gemm_nn(const float* __restrict__ A, int lda, long long zA,
          const float* __restrict__ B, int ldb, long long zB,
          const float* __restrict__ bias,
          float* __restrict__ Cc, int ldc, long long zC, int K) {
  A += (long long)blockIdx.z * zA;
  B += (long long)blockIdx.z * zB;
  Cc += (long long)blockIdx.z * zC;
  const int n0 = blockIdx.x * 64, m0 = blockIdx.y * 64;
  __shared__ __align__(16) float As[2][TILE_F];   // [m][k], pitch AP
  __shared__ __align__(16) float Bs[2][TILE_F];   // [n][k], pitch AP
  const int tid = threadIdx.x;
  const int lane = tid & 31, wave = tid >> 5;
  const int wm = (wave >> 1) * 32, wn = (wave & 1) * 32;
  const int lh = lane >> 4, lm = lane & 15;
  // A staging: thread -> row ar, cols ac..ac+7 (two 16B chunks, direct copy)
  const int ar = tid >> 1, ac = (tid & 1) * 8;
  // B staging: thread -> row (k) br, cols bc..bc+7 (transposed into Bs)
  const int br = tid >> 3, bc = (tid & 7) * 8;
  v8f acc[2][2] = {};
  float4 p0, p1;

  // prologue: stage tile 0
  {
    const float* sa = A + (long long)(m0 + ar) * lda + ac;
    cp16(sa, &As[0][ar * AP + ac]);
    cp16(sa + 4, &As[0][ar * AP + ac + 4]);
    const float4* sb = (const float4*)(B + (long long)br * ldb + n0 + bc);
    p0 = sb[0]; p1 = sb[1];
    float* d = &Bs[0][bc * AP + br];
    d[0 * AP] = p0.x; d[1 * AP] = p0.y; d[2 * AP] = p0.z; d[3 * AP] = p0.w;
    d[4 * AP] = p1.x; d[5 * AP] = p1.y; d[6 * AP] = p1.z; d[7 * AP] = p1.w;
  }
  async_drain();
  __syncthreads();

  int buf = 0;
  for (int k0 = 0; k0 < K; k0 += 16) {
    const int nb = buf ^ 1;
    const bool more = (k0 + 16) < K;
    if (more) {   // issue next tile: async A copy + global loads for B
      const float* sa = A + (long long)(m0 + ar) * lda + (k0 + 16) + ac;
      cp16(sa, &As[nb][ar * AP + ac]);
      cp16(sa + 4, &As[nb][ar * AP + ac + 4]);
      const float4* sb = (const float4*)(B + (long long)(k0 + 16 + br) * ldb + n0 + bc);
      p0 = sb[0]; p1 = sb[1];
    }
    wmma_tile16(As[buf], Bs[buf], wm, wn, lh, lm, acc);
    if (more) {
      float* d = &Bs[nb][bc * AP + br];
      d[0 * AP] = p0.x; d[1 * AP] = p0.y; d[2 * AP] = p0.z; d[3 * AP] = p0.w;
      d[4 * AP] = p1.x; d[5 * AP] = p1.y; d[6 * AP] = p1.z; d[7 * AP] = p1.w;
    }
    async_drain();
    __syncthreads();
    buf = nb;
  }

  #pragma unroll
  for (int i = 0; i < 2; ++i)
    #pragma unroll
    for (int j = 0; j < 2; ++j) {
      int n = n0 + wn + j * 16 + lm;
      float bv = HAS_BIAS ? bias[n] : 0.f;
      int mb = m0 + wm + i * 16 + 8 * lh;   // VGPR r -> row r (+8 for upper half)
      #pragma unroll
      for (int r = 0; r < 8; ++r) {
        float v = acc[i][j][r] + bv;
        if (RELU) v = fmaxf(v, 0.f);
        Cc[(long long)(mb + r) * ldc + n] = v;
      }
    }
}

// ------------------------------------------------------------------
// scores: C[m,n] += scale * (A @ Bt^T)  (Q @ K^T per head via grid.z)
// Both tiles are direct row copies -> fully async staging.
// ------------------------------------------------------------------
__global__ void __launch_bounds__(128)
k_gemm_nt_scores(const float* __restrict__ A, int lda, long long zA,
                 const float* __restrict__ Bt, int ldb, long long zB,
                 float* __restrict__ Cc, int ldc, long long zC,
                 int K, float scale) {
  A += (long long)blockIdx.z * zA;
  Bt += (long long)blockIdx.z * zB;
  Cc += (long long)blockIdx.z * zC;
  const int n0 = blockIdx.x * 64, m0 = blockIdx.y * 64;
  __shared__ __align__(16) float As[2][TILE_F];   // [m][k]
  __shared__ __align__(16) float Bs[2][TILE_F];   // [n][k]
  const int tid = threadIdx.x;
  const int lane = tid & 31, wave = tid >> 5;
  const int wm = (wave >> 1) * 32, wn = (wave & 1) * 32;
  const int lh = lane >> 4, lm = lane & 15;
  const int ar = tid >> 1, ac = (tid & 1) * 8;    // shared by both tiles
  v8f acc[2][2] = {};

  {
    const float* sa = A + (long long)(m0 + ar) * lda + ac;
    cp16(sa, &As[0][ar * AP + ac]);
    cp16(sa + 4, &As[0][ar * AP + ac + 4]);
    const float* sb = Bt + (long long)(n0 + ar) * ldb + ac;
    cp16(sb, &Bs[0][ar * AP + ac]);
    cp16(sb + 4, &Bs[0][ar * AP + ac + 4]);
  }
  async_drain();
  __syncthreads();

  int buf = 0;
  for (int k0 = 0; k0 < K; k0 += 16) {
    const int nb = buf ^ 1;
    const bool more = (k0 + 16) < K;
    if (more) {
      const float* sa = A + (long long)(m0 + ar) * lda + (k0 + 16) + ac;
      cp16(sa, &As[nb][ar * AP + ac]);
      cp16(sa + 4, &As[nb][ar * AP + ac + 4]);
      const float* sb = Bt + (long long)(n0 + ar) * ldb + (k0 + 16) + ac;
      cp16(sb, &Bs[nb][ar * AP + ac]);
      cp16(sb + 4, &Bs[nb][ar * AP + ac + 4]);
    }
    wmma_tile16(As[buf], Bs[buf], wm, wn, lh, lm, acc);
    async_drain();
    __syncthreads();
    buf = nb;
  }

  #pragma unroll
  for (int i = 0; i < 2; ++i)
    #pragma unroll
    for (int j = 0; j < 2; ++j) {
      int n = n0 + wn + j * 16 + lm;
      int mb = m0 + wm + i * 16 + 8 * lh;
      #pragma unroll
      for (int r = 0; r < 8; ++r) {
        long long off = (long long)(mb + r) * ldc + n;
        Cc[off] = fmaf(acc[i][j][r], scale, Cc[off]);   // bias was pre-filled
      }
    }
}

// ------------------------------------------------------------------
// row softmax over j (in place), one block per (i, h)
// ------------------------------------------------------------------
__global__ void __launch_bounds__(256)
k_softmax(float* __restrict__ att) {
  __shared__ float red[256];
  int i = blockIdx.x, h = blockIdx.y, t = threadIdx.x;
  float* row = att + (long long)h * S_ * S_ + (long long)i * S_;
  float v0 = row[t], v1 = row[t + 256], v2 = row[t + 512], v3 = row[t + 768];
  float mx = fmaxf(fmaxf(v0, v1), fmaxf(v2, v3));
  red[t] = mx; __syncthreads();
  for (int s = 128; s > 0; s >>= 1) {
    if (t < s) red[t] = fmaxf(red[t], red[t + s]);
    __syncthreads();
  }
  mx = red[0]; __syncthreads();
  v0 = expf(v0 - mx); v1 = expf(v1 - mx); v2 = expf(v2 - mx); v3 = expf(v3 - mx);
  red[t] = v0 + v1 + v2 + v3; __syncthreads();
  for (int s = 128; s > 0; s >>= 1) {
    if (t < s) red[t] += red[t + s];
    __syncthreads();
  }
  float inv = 1.f / red[0];
  row[t] = v0 * inv; row[t + 256] = v1 * inv;
  row[t + 512] = v2 * inv; row[t + 768] = v3 * inv;
}

// ------------------------------------------------------------------
// x = LayerNorm(x + y) * g + b   (one block per row, D=512)
// ------------------------------------------------------------------
__global__ void __launch_bounds__(256)
k_add_ln(float* __restrict__ x, const float* __restrict__ y,
         const float* __restrict__ g, const float* __restrict__ b) {
  __shared__ float red[256];
  int i = blockIdx.x, t = threadIdx.x;
  float* row = x + (long long)i * D_;
  const float* yr = y + (long long)i * D_;
  float a0 = row[t] + yr[t];
  float a1 = row[t + 256] + yr[t + 256];
  red[t] = a0 + a1; __syncthreads();
  for (int s = 128; s > 0; s >>= 1) {
    if (t < s) red[t] += red[t + s];
    __syncthreads();
  }
  float mu = red[0] * (1.f / D_); __syncthreads();
  float d0 = a0 - mu, d1 = a1 - mu;
  red[t] = d0 * d0 + d1 * d1; __syncthreads();
  for (int s = 128; s > 0; s >>= 1) {
    if (t < s) red[t] += red[t + s];
    __syncthreads();
  }
  float rs = rsqrtf(red[0] * (1.f / D_) + EPS_);
  row[t] = d0 * rs * g[t] + b[t];
  row[t + 256] = d1 * rs * g[t + 256] + b[t + 256];
}

// ------------------------------------------------------------------
// mean pool over sequence
// ------------------------------------------------------------------
__global__ void __launch_bounds__(256)
k_pool(const float* __restrict__ x, float* __restrict__ pooled) {
  int d = blockIdx.x * 256 + threadIdx.x;
  float s = 0.f;
  for (int i = 0; i < S_; ++i) s += x[(long long)i * D_ + d];
  pooled[d] = s * (1.f / S_);
}

// ------------------------------------------------------------------
// classifier: relu(pooled @ c1w + c1b) @ c2w + c2b  (single block)
// ------------------------------------------------------------------
__global__ void __launch_bounds__(256)
k_cls(const float* __restrict__ pooled, const float* __restrict__ c1w,
      const float* __restrict__ c1b, const float* __restrict__ c2w,
      const float* __restrict__ c2b, float* __restrict__ out) {
  __shared__ float hid[D_ / 2];
  int t = threadIdx.x;
  float a = c1b[t];
  for (int k = 0; k < D_; ++k) a = fmaf(pooled[k], c1w[k * (D_ / 2) + t], a);
  hid[t] = fmaxf(a, 0.f);
  __syncthreads();
  if (t < NC_) {
    float o = c2b[t];
    for (int k = 0; k < D_ / 2; ++k) o = fmaf(hid[k], c2w[k * NC_ + t], o);
    out[t] = o;
  }
}

// ------------------------------------------------------------------
extern "C" void kernel_launch(void* const* d_in, const int* in_sizes, int n_in,
                              void* d_out, int out_size, void* d_ws, size_t ws_size,
                              hipStream_t stream) {
  const float* features = (const float*)d_in[0];
  const float* positions = (const float*)d_in[1];
  const float* freq_bands = (const float*)d_in[2];
  const float* in_w = (const float*)d_in[3];
  const float* in_b = (const float*)d_in[4];
  const float* qw = (const float*)d_in[5];
  const float* qb = (const float*)d_in[6];
  const float* kw = (const float*)d_in[7];
  const float* kb = (const float*)d_in[8];
  const float* vw = (const float*)d_in[9];
  const float* vb = (const float*)d_in[10];
  const float* ow = (const float*)d_in[11];
  const float* ob = (const float*)d_in[12];
  const float* db1w = (const float*)d_in[13];
  const float* db1b = (const float*)d_in[14];
  const float* db2w = (const float*)d_in[15];
  const float* db2b = (const float*)d_in[16];
  const float* n1g = (const float*)d_in[17];
  const float* n1b = (const float*)d_in[18];
  const float* n2g = (const float*)d_in[19];
  const float* n2b = (const float*)d_in[20];
  const float* f1w = (const float*)d_in[21];
  const float* f1b = (const float*)d_in[22];
  const float* f2w = (const float*)d_in[23];
  const float* f2b = (const float*)d_in[24];
  const float* c1w = (const float*)d_in[25];
  const float* c1b = (const float*)d_in[26];
  const float* c2w = (const float*)d_in[27];
  const float* c2b = (const float*)d_in[28];
  float* out = (float*)d_out;

  const long long SD = (long long)S_ * D_;
  const long long SS = (long long)S_ * S_;
  float* w = (float*)d_ws;
  float* x = w;                      // S*D
  float* dist = x + SD;              // S*S
  float* q = dist + SS;              // S*D
  float* kbuf = q + SD;              // S*D
  float* v = kbuf + SD;              // S*D
  float* att = v + SD;               // H*S*S
  float* ao = att + (long long)H_ * SS;  // S*D
  float* tmp = ao + SD;              // S*D
  float* ffh = tmp + SD;             // S*DFF
  float* pooled = ffh + (long long)S_ * DFF_;  // D

  const float scale = 0.125f;        // 1/sqrt(64)

  k_input_pe<<<(S_ * D_) / 256, 256, 0, stream>>>(features, positions, freq_bands,
                                                  in_w, in_b, x);
  k_dist<<<(S_ * S_) / 256, 256, 0, stream>>>(positions, dist);

  for (int l = 0; l < L_; ++l) {
    // distance-bias pre-fill of score buffer
    k_bias<<<(S_ * S_) / 256, 256, 0, stream>>>(dist, db1w + l * DB_, db1b + l * DB_,
                                                db2w + l * DB_ * H_, db2b + l * H_, att);
    // Q, K, V projections (M=1024, N=512, K=512)
    k_gemm_nn<false, true><<<dim3(8, 16, 1), 128, 0, stream>>>(
        x, D_, 0, qw + (long long)l * D_ * D_, D_, 0, qb + l * D_, q, D_, 0, D_);
    k_gemm_nn<false, true><<<dim3(8, 16, 1), 128, 0, stream>>>(
        x, D_, 0, kw + (long long)l * D_ * D_, D_, 0, kb + l * D_, kbuf, D_, 0, D_);
    k_gemm_nn<false, true><<<dim3(8, 16, 1), 128, 0, stream>>>(
        x, D_, 0, vw + (long long)l * D_ * D_, D_, 0, vb + l * D_, v, D_, 0, D_);
    // scores += scale * Q_h @ K_h^T  (per head via grid.z)
    k_gemm_nt_scores<<<dim3(16, 16, H_), 128, 0, stream>>>(
        q, D_, HD_, kbuf, D_, HD_, att, S_, SS, HD_, scale);
    // softmax over j
    k_softmax<<<dim3(S_, H_), 256, 0, stream>>>(att);
    // attn @ V  (M=1024, N=64 per head, K=1024)
    k_gemm_nn<false, false><<<dim3(1, 16, H_), 128, 0, stream>>>(
        att, S_, SS, v, D_, HD_, nullptr, ao, D_, HD_, S_);
    // output projection
    k_gemm_nn<false, true><<<dim3(8, 16, 1), 128, 0, stream>>>(
        ao, D_, 0, ow + (long long)l * D_ * D_, D_, 0, ob + l * D_, tmp, D_, 0, D_);
    k_add_ln<<<S_, 256, 0, stream>>>(x, tmp, n1g + l * D_, n1b + l * D_);
    // FFN
    k_gemm_nn<true, true><<<dim3(32, 16, 1), 128, 0, stream>>>(
        x, D_, 0, f1w + (long long)l * D_ * DFF_, DFF_, 0, f1b + l * DFF_,
        ffh, DFF_, 0, D_);
    k_gemm_nn<false, true><<<dim3(8, 16, 1), 128, 0, stream>>>(
        ffh, DFF_, 0, f2w + (long long)l * DFF_ * D_, D_, 0, f2b + l * D_,
        tmp, D_, 0, DFF_);
    k_add_ln<<<S_, 256, 0, stream>>>(x, tmp, n2g + l * D_, n2b + l * D_);
  }

  k_pool<<<D_ / 256, 256, 0, stream>>>(x, pooled);
  k_cls<<<1, 256, 0, stream>>>(pooled, c1w, c1b, c2w, c2b, out);
}